// QuantumLayer_4664334483485
// MI455X (gfx1250) — compile-verified
//
#include <hip/hip_runtime.h>

// MI455X / gfx1250, wave32. One wave simulates one batch row's 8-qubit state
// (256 complex amps = 16 f32 VGPRs/lane). Post-GEMM (qz @ w_post.T + b) done
// with V_WMMA_F32_16X16X4_F32 (fp32 WMMA, K=8 split into 2 chained K=4 ops).

typedef __attribute__((ext_vector_type(2))) float v2f;
typedef __attribute__((ext_vector_type(8))) float v8f;

#define DEVINL __device__ __forceinline__

DEVINL float shflx(float v, int mask) { return __shfl_xor(v, mask, 32); }

__global__ __launch_bounds__(512)
void qcircuit_wmma_kernel(const float* __restrict__ x,
                          const float* __restrict__ weights,
                          const float* __restrict__ w_post,
                          const float* __restrict__ b_post,
                          float* __restrict__ out)
{
    __shared__ float gc[16][8];   // fused per-(layer,qubit) 2x2 complex gates
    __shared__ float qz[16][8];   // per-row <Z_w> staging for WMMA

    const int tid  = threadIdx.x;
    const int lane = tid & 31;
    const int wid  = tid >> 5;    // 16 waves -> 16 batch rows per block

    // ---- Phase 0: build fused U = Rz*Ry*Rx for all 16 (layer,qubit) gates ----
    if (tid < 16) {
        const int l = tid >> 3, q = tid & 7;
        const float* wl = weights + l * 24 + q * 3;
        float sx, cx, sy, cy, sz, cz;
        sincosf(0.5f * wl[0], &sx, &cx);
        sincosf(0.5f * wl[1], &sy, &cy);
        sincosf(0.5f * wl[2], &sz, &cz);
        // M = Ry*Rx
        const float m00r =  cy*cx, m00i =  sy*sx;
        const float m01r = -sy*cx, m01i = -cy*sx;
        const float m10r =  sy*cx, m10i = -cy*sx;
        const float m11r =  cy*cx, m11i = -sy*sx;
        // row0 *= (cz - i sz), row1 *= (cz + i sz)
        gc[tid][0] = m00r*cz + m00i*sz;  gc[tid][1] = m00i*cz - m00r*sz;
        gc[tid][2] = m01r*cz + m01i*sz;  gc[tid][3] = m01i*cz - m01r*sz;
        gc[tid][4] = m10r*cz - m10i*sz;  gc[tid][5] = m10i*cz + m10r*sz;
        gc[tid][6] = m11r*cz - m11i*sz;  gc[tid][7] = m11i*cz + m11r*sz;
    }
    __syncthreads();

    const int b = blockIdx.x * 16 + wid;

    // ---- Initial product state from RY(x[b,0..7]) on |0..0> ----
    // amp(a) = prod_w (bit_{7-w}(a) ? sin(x_w/2) : cos(x_w/2)); a = lane<<3 | r
    float myang = (lane < 8) ? x[(size_t)b * 128 + lane] : 0.0f;
    float sl, cl;
    sincosf(0.5f * myang, &sl, &cl);
    float cw[8], sw[8];
    #pragma unroll
    for (int w = 0; w < 8; ++w) { cw[w] = __shfl(cl, w, 32); sw[w] = __shfl(sl, w, 32); }

    const float lp = ((lane & 16) ? sw[0] : cw[0]) * ((lane & 8) ? sw[1] : cw[1])
                   * ((lane & 4)  ? sw[2] : cw[2]) * ((lane & 2) ? sw[3] : cw[3])
                   * ((lane & 1)  ? sw[4] : cw[4]);
    float re[8], im[8];
    #pragma unroll
    for (int r = 0; r < 8; ++r) {
        const float f = ((r & 4) ? sw[5] : cw[5]) * ((r & 2) ? sw[6] : cw[6])
                      * ((r & 1) ? sw[7] : cw[7]);
        re[r] = lp * f;
        im[r] = 0.0f;
    }

    // ---- 2 layers: 8 fused 1q gates + CNOT ladder ----
    #pragma unroll
    for (int l = 0; l < 2; ++l) {
        #pragma unroll
        for (int q = 0; q < 8; ++q) {
            const float* u = gc[l * 8 + q];
            const float u00r = u[0], u00i = u[1], u01r = u[2], u01i = u[3];
            const float u10r = u[4], u10i = u[5], u11r = u[6], u11i = u[7];
            if (q < 5) {
                // cross-lane gate: qubit q <-> amp bit (7-q) <-> lane bit (4-q)
                const int mask = 16 >> q;
                const bool hi = (lane & mask) != 0;
                const float ar = hi ? u11r : u00r, ai = hi ? u11i : u00i;
                const float br = hi ? u10r : u01r, bi = hi ? u10i : u01i;
                #pragma unroll
                for (int r = 0; r < 8; ++r) {
                    const float ore = shflx(re[r], mask), oim = shflx(im[r], mask);
                    const float nr = ar*re[r] - ai*im[r] + br*ore - bi*oim;
                    const float ni = ar*im[r] + ai*re[r] + br*oim + bi*ore;
                    re[r] = nr; im[r] = ni;
                }
            } else {
                // in-register gate: qubit 5/6/7 <-> reg bit 2/1/0
                const int rb = (q == 5) ? 4 : ((q == 6) ? 2 : 1);
                #pragma unroll
                for (int r = 0; r < 8; ++r) {
                    if (r & rb) continue;
                    const int r1 = r | rb;
                    const float lre = re[r],  lim = im[r];
                    const float hre = re[r1], him = im[r1];
                    re[r]  = u00r*lre - u00i*lim + u01r*hre - u01i*him;
                    im[r]  = u00r*lim + u00i*lre + u01r*him + u01i*hre;
                    re[r1] = u10r*lre - u10i*lim + u11r*hre - u11i*him;
                    im[r1] = u10r*lim + u10i*lre + u11r*him + u11i*hre;
                }
            }
        }
        // CNOT(q,q+1) for q=0..3: ctrl lane bit (4-q), target lane bit (3-q)
        #pragma unroll
        for (int k = 0; k < 4; ++k) {
            const int mask = 8 >> k;     // 8,4,2,1
            const int ctrl = 16 >> k;    // 16,8,4,2
            const bool c = (lane & ctrl) != 0;
            #pragma unroll
            for (int r = 0; r < 8; ++r) {
                const float tr = shflx(re[r], mask), ti = shflx(im[r], mask);
                re[r] = c ? tr : re[r];
                im[r] = c ? ti : im[r];
            }
        }
        // CNOT(4,5): ctrl lane bit0, target reg bit2 -> predicated reg swap
        {
            const bool c = (lane & 1) != 0;
            #pragma unroll
            for (int r = 0; r < 4; ++r) {
                float t0 = re[r], t1 = re[r + 4];
                re[r] = c ? t1 : t0;  re[r + 4] = c ? t0 : t1;
                float s0 = im[r], s1 = im[r + 4];
                im[r] = c ? s1 : s0;  im[r + 4] = c ? s0 : s1;
            }
        }
        // CNOT(5,6): swap regs (4,6),(5,7)
        { float t;
          t = re[4]; re[4] = re[6]; re[6] = t;  t = re[5]; re[5] = re[7]; re[7] = t;
          t = im[4]; im[4] = im[6]; im[6] = t;  t = im[5]; im[5] = im[7]; im[7] = t; }
        // CNOT(6,7): swap regs (2,3),(6,7)
        { float t;
          t = re[2]; re[2] = re[3]; re[3] = t;  t = re[6]; re[6] = re[7]; re[7] = t;
          t = im[2]; im[2] = im[3]; im[3] = t;  t = im[6]; im[6] = im[7]; im[7] = t; }
        // CNOT(7,0): ctrl reg bit0, target lane bit4 -> swap odd regs across lane^16
        #pragma unroll
        for (int r = 1; r < 8; r += 2) {
            re[r] = shflx(re[r], 16);
            im[r] = shflx(im[r], 16);
        }
    }

    // ---- Measurement: qz[w] = sum_a |amp|^2 * (1 - 2*bit_{7-w}(a)) ----
    float ptot = 0.f, s5 = 0.f, s6 = 0.f, s7 = 0.f;
    #pragma unroll
    for (int r = 0; r < 8; ++r) {
        const float p = re[r]*re[r] + im[r]*im[r];
        ptot += p;
        s5 += (r & 4) ? -p : p;
        s6 += (r & 2) ? -p : p;
        s7 += (r & 1) ? -p : p;
    }
    float v[8];
    v[0] = (lane & 16) ? -ptot : ptot;
    v[1] = (lane & 8)  ? -ptot : ptot;
    v[2] = (lane & 4)  ? -ptot : ptot;
    v[3] = (lane & 2)  ? -ptot : ptot;
    v[4] = (lane & 1)  ? -ptot : ptot;
    v[5] = s5; v[6] = s6; v[7] = s7;
    #pragma unroll
    for (int k = 0; k < 8; ++k) {
        float t = v[k];
        t += shflx(t, 1); t += shflx(t, 2); t += shflx(t, 4);
        t += shflx(t, 8); t += shflx(t, 16);
        v[k] = t;
    }
    if (lane == 0) {
        #pragma unroll
        for (int k = 0; k < 8; ++k) qz[wid][k] = v[k];
    }
    __syncthreads();

    // ---- WMMA post-GEMM: out[16x128 tile] = qz[16x8] @ w_post^T[8x128] + b ----
    // A 16x4 f32 layout: lanes 0-15 -> M=0..15; VGPR0 = K {0|2}, VGPR1 = K {1|3}
    if (wid < 8) {
        const int m  = lane & 15;
        const int kb = (lane >> 4) * 2;
        v2f a0, a1, b0, b1;
        a0.x = qz[m][kb];     a0.y = qz[m][kb + 1];
        a1.x = qz[m][4 + kb]; a1.y = qz[m][5 + kb];
        const int col = wid * 16 + m;           // output feature / B-matrix N
        const float* wp = w_post + col * 8;     // w_post[col][k] = B[k][col]
        b0.x = wp[kb];     b0.y = wp[kb + 1];
        b1.x = wp[4 + kb]; b1.y = wp[5 + kb];
        v8f acc = {};
        acc = __builtin_amdgcn_wmma_f32_16x16x4_f32(false, a0, false, b0,
                                                    (short)0, acc, false, false);
        acc = __builtin_amdgcn_wmma_f32_16x16x4_f32(false, a1, false, b1,
                                                    (short)0, acc, false, false);
        const float bias = b_post[col];
        const int row0 = blockIdx.x * 16 + (lane >> 4) * 8;
        #pragma unroll
        for (int j = 0; j < 8; ++j)
            out[(size_t)(row0 + j) * 128 + col] = acc[j] + bias;
    }
}

extern "C" void kernel_launch(void* const* d_in, const int* in_sizes, int n_in,
                              void* d_out, int out_size, void* d_ws, size_t ws_size,
                              hipStream_t stream) {
    const float* x       = (const float*)d_in[0];   // (B,128) f32
    const float* weights = (const float*)d_in[1];   // (2,8,3) f32
    const float* w_post  = (const float*)d_in[2];   // (128,8) f32
    const float* b_post  = (const float*)d_in[3];   // (128,)  f32
    float* out = (float*)d_out;                     // (B,128) f32

    const int B = in_sizes[0] / 128;                // 32768
    const int blocks = B / 16;                      // 16 rows (waves) per block
    hipLaunchKernelGGL(qcircuit_wmma_kernel, dim3(blocks), dim3(512), 0, stream,
                       x, weights, w_post, b_post, out);
}